// OrNet_66915590472439
// MI455X (gfx1250) — compile-verified
//
#include <hip/hip_runtime.h>

// OrNet resonate-and-fire: collapse B=32768 neurons onto the 3 distinct input
// classes, scan T=1000 steps once per class (Phase A), then gather per-neuron
// results as a one-hot x traj matmul via V_WMMA_F32_16X16X4_F32 (Phase B).

#define T_STEPS   1000
#define FREQ_ON   64
#define FREQ_OFF  32
#define SCALE_V   20.0f
#define DT_V      0.01f

typedef __attribute__((ext_vector_type(2))) float v2f;
typedef __attribute__((ext_vector_type(8))) float v8f;

// ---------------- Phase A: 3-class trajectory scan (1 wave) ----------------
__global__ void ornet_scan(const float* __restrict__ params,
                           float* __restrict__ traj /* 4 floats in d_ws */) {
    const int lane = threadIdx.x;
    const float beta = params[0];
    const float freq = params[1];
    const int   c    = (lane < 3) ? lane : 0;      // class = number of "on" inputs
    const float cn   = (float)c;                   // count of on-trains
    const float cf   = (float)(2 - c);             // count of off-trains

    float mr = 0.0f, mi = 0.0f;
    for (int t = 0; t < T_STEPS; ++t) {
        const float on  = ((t >= FREQ_ON)  && ((t & (FREQ_ON  - 1)) == 0)) ? SCALE_V : 0.0f;
        const float off = ((t >= FREQ_OFF) && ((t & (FREQ_OFF - 1)) == 0)) ? SCALE_V : 0.0f;
        const float z   = cn * on + cf * off;

        float mr_n = mr + DT_V * (beta * mr - freq * mi + z);
        float mi_n = mi + DT_V * (freq * mr + beta * mi);
        if (mi_n > 0.0f) mr_n = 0.0f;              // spike -> reset real part
        mr = mr_n;
        mi = mi_n;
    }
    if (lane < 3) traj[lane] = mi;
    if (lane == 3) traj[3] = 0.0f;                 // padding row K=3
}

// ------------- Phase B: WMMA one-hot gather, 16 neurons per wave -------------
// D = A(16x4: traj broadcast over rows) x B(4x16: per-neuron one-hot columns).
// Every column of D equals traj[class(n)], exactly (products are 0 or traj[k]).
__global__ void ornet_gather(const float* __restrict__ x1,
                             const float* __restrict__ x2,
                             const float* __restrict__ traj,
                             float* __restrict__ out, int B) {
    const int gid  = blockIdx.x * blockDim.x + threadIdx.x;
    const int wave = gid >> 5;
    const int lane = threadIdx.x & 31;
    const int base = wave * 16;

    const int n  = base + (lane & 15);
    const int nc = (n < B) ? n : (B - 1);          // clamp: keeps EXEC uniform
    const int cls = (x1[nc] > 0.0f ? 1 : 0) + (x2[nc] > 0.0f ? 1 : 0);

    const float t0 = traj[0];
    const float t1 = traj[1];
    const float t2 = traj[2];

    const bool lo = (lane < 16);
    // A 16x4 f32 (documented layout): lanes0-15 -> {K0,K1}, lanes16-31 -> {K2,K3}
    v2f a;
    a.x = lo ? t0 : t2;
    a.y = lo ? t1 : 0.0f;
    // B 4x16 f32 (mirror layout): lanes0-15 -> rows {K0,K1} at N=lane,
    // lanes16-31 -> rows {K2,K3} at N=lane-16. One-hot of this neuron's class.
    v2f b;
    b.x = lo ? (cls == 0 ? 1.0f : 0.0f) : (cls == 2 ? 1.0f : 0.0f);
    b.y = lo ? (cls == 1 ? 1.0f : 0.0f) : 0.0f;

    v8f cacc = {};
    // (neg_a, A, neg_b, B, c_mod, C, reuse_a, reuse_b)
    v8f d = __builtin_amdgcn_wmma_f32_16x16x4_f32(
        false, a, false, b, (short)0, cacc, false, false);

    // Lane l<16 holds D[M=r, N=l] in d[r]; all columns identical per neuron,
    // so d[0] on lane l is out[base+l]. Coalesced 16-lane store.
    if (lo && n < B) out[n] = d[0];
}

extern "C" void kernel_launch(void* const* d_in, const int* in_sizes, int n_in,
                              void* d_out, int out_size, void* d_ws, size_t ws_size,
                              hipStream_t stream) {
    const float* x1     = (const float*)d_in[0];
    const float* x2     = (const float*)d_in[1];
    const float* params = (const float*)d_in[2];
    float*       out    = (float*)d_out;
    float*       traj   = (float*)d_ws;           // 4 floats of scratch
    const int    B      = in_sizes[0];

    // Phase A: one wave, sequential 1000-step scan of the 3 classes.
    ornet_scan<<<1, 32, 0, stream>>>(params, traj);

    // Phase B: one 16-neuron tile per wave32.
    const int nwaves  = (B + 15) / 16;
    const int threads = nwaves * 32;
    const int block   = 256;
    const int grid    = (threads + block - 1) / block;
    ornet_gather<<<grid, block, 0, stream>>>(x1, x2, traj, out, B);
}